// GliaShrink_60662118089241
// MI455X (gfx1250) — compile-verified
//
#include <hip/hip_runtime.h>

// Skinny GEMM: out[:, :3] = X @ W[:3].T + IN*bias[:3], rest of out = 0.
// Memory bound (~131 MB @ 23.3 TB/s). Split-K WMMA f32 16x16x4 + NT zero fill.

typedef float v2f __attribute__((ext_vector_type(2)));
typedef float v4f __attribute__((ext_vector_type(4)));
typedef float v8f __attribute__((ext_vector_type(8)));

#define KC 64                   // K-chunk staged in LDS per wave
#define LDS_STRIDE 66           // KC + 2 pad: kills stride-64 bank conflicts, keeps 8B align
#define ROWS_PER_WAVE 16        // WMMA M
#define WAVES_PER_BLOCK 8
#define ROWS_PER_BLOCK (ROWS_PER_WAVE * WAVES_PER_BLOCK)

// ---------------------------------------------------------------- zero fill
__global__ void __launch_bounds__(256)
fill_zero_kernel(float* __restrict__ out, long n) {
  long i = ((long)blockIdx.x * blockDim.x + threadIdx.x) * 4;
  if (i + 3 < n) {
    v4f z = {0.f, 0.f, 0.f, 0.f};
    __builtin_nontemporal_store(z, (v4f*)(out + i));
  } else {
    for (long j = i; j < n; ++j) out[j] = 0.f;
  }
}

// ---------------------------------------------------------------- WMMA GEMM
// grid.x = row tiles (128 rows each), grid.y = K splits.
// finalWrite==1: write out[row*OUT + n] = c + IN*bias[n]   (KSPLIT==1 path)
// finalWrite==0: write partial[(split*nRows + row)*4 + n] = c
__global__ void __launch_bounds__(256)
skinny_gemm_wmma(const float* __restrict__ x, const float* __restrict__ w,
                 const float* __restrict__ bias,
                 float* __restrict__ partial, float* __restrict__ out,
                 int IN, int OUT, int kLen, int nRows, int finalWrite)
{
  extern __shared__ float lds[];
  float* lds_w = lds;                                   // wcols * kLen floats
  const int tid  = threadIdx.x;
  const int lane = tid & 31;
  const int wave = tid >> 5;
  float* lds_a = lds + 3 * kLen + wave * (ROWS_PER_WAVE * LDS_STRIDE);

  const int  split   = blockIdx.y;
  const long kBase   = (long)split * kLen;
  const int  rowBase = blockIdx.x * ROWS_PER_BLOCK + wave * ROWS_PER_WAVE;
  const int  wcols   = (OUT < 3) ? OUT : 3;

  // Stage the live weight rows for this K-slice (shared by all 8 waves).
  for (int i = tid; i < wcols * kLen; i += blockDim.x) {
    int nrow = i / kLen;
    int k    = i - nrow * kLen;
    lds_w[nrow * kLen + k] = w[(long)nrow * IN + kBase + k];
  }
  __syncthreads();

  const int   m      = lane & 15;          // N (cols) / M (rows) sub-index
  const int   khalf  = lane >> 4;          // lane half selects K pair
  const int   nclamp = (m < wcols) ? m : 0;
  const bool  live   = (m < wcols);

  v8f c = {};                              // 16x16 f32 accumulator

  for (int kc = 0; kc < kLen; kc += KC) {
    // Coalesced NT stream of the 16 x KC input tile into padded LDS.
    #pragma unroll
    for (int it = 0; it < (ROWS_PER_WAVE * KC / 4) / 32; ++it) {
      int f  = it * 32 + lane;
      int r  = f >> 4;                     // KC/4 = 16 float4 per row
      int c4 = f & 15;
      const v4f v = __builtin_nontemporal_load(
          (const v4f*)(x + (long)(rowBase + r) * IN + kBase + kc + c4 * 4));
      float* dst = lds_a + r * LDS_STRIDE + c4 * 4;
      v2f lo; lo.x = v.x; lo.y = v.y;
      v2f hi; hi.x = v.z; hi.y = v.w;
      *(v2f*)(dst)     = lo;
      *(v2f*)(dst + 2) = hi;
    }
    __syncthreads();

    // 16 WMMA steps per chunk. A layout (ISA 7.12.2, 32-bit A 16x4):
    //   lane<16: M=lane, K=0,1 ; lane>=16: M=lane-16, K=2,3
    #pragma unroll
    for (int kk = 0; kk < KC; kk += 4) {
      const v2f av = *(const v2f*)(lds_a + m * LDS_STRIDE + kk + 2 * khalf);
      const v2f bv = *(const v2f*)(lds_w + nclamp * kLen + kc + kk + 2 * khalf);
      v2f a = av;
      v2f b; b.x = live ? bv.x : 0.f;  b.y = live ? bv.y : 0.f;
      c = __builtin_amdgcn_wmma_f32_16x16x4_f32(
          /*neg_a=*/false, a, /*neg_b=*/false, b,
          /*c_mod=*/(short)0, c, /*reuse_a=*/false, /*reuse_b=*/false);
    }
    __syncthreads();
  }

  // C/D layout: vgpr i, lane l -> M = i + 8*(l>=16), N = l%16
  if (finalWrite) {
    if (m < wcols) {
      float bv = (float)IN * bias[m];
      #pragma unroll
      for (int i = 0; i < 8; ++i) {
        int row = rowBase + i + 8 * khalf;
        out[(long)row * OUT + m] = c[i] + bv;
      }
    }
  } else {
    if (m < wcols) {
      #pragma unroll
      for (int i = 0; i < 8; ++i) {
        int row = rowBase + i + 8 * khalf;
        partial[((long)split * nRows + row) * 4 + m] = c[i];
      }
    }
  }
}

// ---------------------------------------------------------------- reduction
__global__ void __launch_bounds__(256)
reduce_bias_kernel(const float* __restrict__ partial, const float* __restrict__ bias,
                   float* __restrict__ out, int nRows, int OUT, int IN,
                   int ksplit, int wcols)
{
  int idx = blockIdx.x * blockDim.x + threadIdx.x;
  if (idx >= nRows * wcols) return;
  int r = idx / wcols;
  int n = idx - r * wcols;
  float s = 0.f;
  for (int sp = 0; sp < ksplit; ++sp)        // fixed order -> deterministic
    s += partial[((long)sp * nRows + r) * 4 + n];
  out[(long)r * OUT + n] = s + (float)IN * bias[n];
}

// ---------------------------------------------------------------- launcher
extern "C" void kernel_launch(void* const* d_in, const int* in_sizes, int n_in,
                              void* d_out, int out_size, void* d_ws, size_t ws_size,
                              hipStream_t stream) {
  const float* x    = (const float*)d_in[0];
  const float* w    = (const float*)d_in[1];
  const float* bias = (const float*)d_in[2];
  float* out = (float*)d_out;

  const int OUT   = in_sizes[2];
  const int IN    = in_sizes[1] / OUT;
  const int B     = in_sizes[0] / IN;
  const int wcols = (OUT < 3) ? OUT : 3;

  // 1) Zero the whole output with streaming NT stores (67 MB dominates).
  {
    long n  = (long)out_size;
    long n4 = (n + 3) / 4;
    int blocks = (int)((n4 + 255) / 256);
    fill_zero_kernel<<<blocks, 256, 0, stream>>>(out, n);
  }

  // 2) Skinny WMMA GEMM; split K 16 ways to put ~4K waves on the machine.
  int KSPLIT = 16;
  if ((IN % (KSPLIT * KC)) != 0 || (B % ROWS_PER_BLOCK) != 0 ||
      ws_size < (size_t)KSPLIT * B * 4 * sizeof(float))
    KSPLIT = 1;
  const int kLen = IN / KSPLIT;
  const size_t ldsBytes =
      (size_t)(3 * kLen + WAVES_PER_BLOCK * ROWS_PER_WAVE * LDS_STRIDE) * sizeof(float);
  dim3 grid(B / ROWS_PER_BLOCK, KSPLIT);

  if (KSPLIT > 1) {
    skinny_gemm_wmma<<<grid, 256, ldsBytes, stream>>>(
        x, w, bias, (float*)d_ws, nullptr, IN, OUT, kLen, B, 0);
    int total = B * wcols;
    reduce_bias_kernel<<<(total + 255) / 256, 256, 0, stream>>>(
        (const float*)d_ws, bias, out, B, OUT, IN, KSPLIT, wcols);
  } else {
    skinny_gemm_wmma<<<grid, 256, ldsBytes, stream>>>(
        x, w, bias, nullptr, out, IN, OUT, kLen, B, 1);
  }
}